// Gemma3nAudioRelativePositionEmbedding_81449759801606
// MI455X (gfx1250) — compile-verified
//
#include <hip/hip_runtime.h>
#include <math.h>

// Shapes (hardcoded from reference)
#define Bc   8
#define Wc   512
#define Uc   12
#define Hc   8
#define Dc   192
#define Kc   24
#define Fc   13
#define CHc  1536      // H*D
#define QSTR 196       // LDS row stride (floats): 192 + 4 pad -> conflict-free fragment reads

typedef float v2f __attribute__((ext_vector_type(2)));
typedef float v8f __attribute__((ext_vector_type(8)));

// ---------------- prologue 1: sinusoidal embedding [F][CH] ----------------
__global__ void sinemb_kernel(float* __restrict__ sinemb) {
    int i = blockIdx.x * blockDim.x + threadIdx.x;
    if (i >= Fc * CHc) return;
    int f = i / CHc, c = i % CHc;
    float pos = (float)(12 - f);                    // MAX_BACK .. 0 descending
    const float log_inc = 9.210340371976184f / 767.0f; // ln(10000)/(num_ts-1)
    int ts = (c < CHc / 2) ? c : (c - CHc / 2);
    float scaled = pos * expf(-(float)ts * log_inc);
    sinemb[i] = (c < CHc / 2) ? sinf(scaled) : cosf(scaled);
}

// ---------------- prologue 2: proj[f][o] = sum_c sinemb[f][c] * pos_w[o][c] ----------------
__global__ void proj_kernel(const float* __restrict__ sinemb,
                            const float* __restrict__ pos_w,
                            float* __restrict__ proj) {
    int i = blockIdx.x * blockDim.x + threadIdx.x;
    if (i >= Fc * CHc) return;
    int f = i / CHc, o = i % CHc;
    const float* se   = sinemb + f * CHc;
    const float* wrow = pos_w + (size_t)o * CHc;
    float acc = 0.0f;
    for (int c = 0; c < CHc; ++c) acc = fmaf(se[c], wrow[c], acc);
    proj[f * CHc + o] = acc;   // layout [F][H*D] -> proj[f][h][d]
}

// ---------------- main: per (b,w,h) tile, WMMA f32 16x16x4 ----------------
__launch_bounds__(128)
__global__ void rel_attn_kernel(const float* __restrict__ q,
                                const float* __restrict__ k,
                                const float* __restrict__ proj,
                                float* __restrict__ out) {
    __shared__ alignas(16) float Qs[16 * QSTR];   // rows 12..15 zero
    __shared__ alignas(16) float Ks[32 * QSTR];   // rows 24..31 zero
    __shared__ alignas(16) float Ps[16 * QSTR];   // rows 13..15 zero
    __shared__ float Sac[16 * 32];                // [u][c] scores (content)
    __shared__ float Sbd[16 * 16];                // [u][f] scores (positional)

    const int blk = blockIdx.x;
    const int h   = blk % Hc;
    const int bw  = blk / Hc;                     // b*W + w

    const float* qg = q    + (size_t)bw * (Uc * Hc * Dc) + h * Dc;
    const float* kg = k    + (size_t)bw * (Kc * Hc * Dc) + h * Dc;
    const float* pg = proj + h * Dc;

    const int t = threadIdx.x;

    // LDS byte offsets of staging buffers (flat-aperture low 32 bits == DS address)
    const unsigned lQ = (unsigned)(size_t)&Qs[0];
    const unsigned lK = (unsigned)(size_t)&Ks[0];
    const unsigned lP = (unsigned)(size_t)&Ps[0];

    // ---- async global->LDS staging of valid rows (row = 48 float4, global row stride = 6144 B) ----
    // Q/K are single-use streams: non-temporal. proj is L2-hot (reused by all blocks): default policy.
    for (int i = t; i < Uc * 48; i += 128) {          // Q: 12 rows
        int r = i / 48, c4 = i % 48;
        unsigned lds = lQ + (unsigned)((r * QSTR + c4 * 4) * 4);
        unsigned gof = (unsigned)(r * 6144 + c4 * 16);
        asm volatile("global_load_async_to_lds_b128 %0, %1, %2 th:TH_LOAD_NT"
                     :: "v"(lds), "v"(gof), "s"(qg) : "memory");
    }
    for (int i = t; i < Kc * 48; i += 128) {          // K: 24 rows
        int r = i / 48, c4 = i % 48;
        unsigned lds = lK + (unsigned)((r * QSTR + c4 * 4) * 4);
        unsigned gof = (unsigned)(r * 6144 + c4 * 16);
        asm volatile("global_load_async_to_lds_b128 %0, %1, %2 th:TH_LOAD_NT"
                     :: "v"(lds), "v"(gof), "s"(kg) : "memory");
    }
    for (int i = t; i < Fc * 48; i += 128) {          // proj: 13 rows
        int r = i / 48, c4 = i % 48;
        unsigned lds = lP + (unsigned)((r * QSTR + c4 * 4) * 4);
        unsigned gof = (unsigned)(r * 6144 + c4 * 16);
        asm volatile("global_load_async_to_lds_b128 %0, %1, %2"
                     :: "v"(lds), "v"(gof), "s"(pg) : "memory");
    }

    // ---- zero-fill pad rows with plain DS stores (disjoint from async destinations) ----
    {
        float4 z = make_float4(0.f, 0.f, 0.f, 0.f);
        for (int i = t; i < 4 * 49; i += 128) {       // Q rows 12..15, full 196-float rows
            int r = 12 + i / 49, c = i % 49;
            *(float4*)&Qs[r * QSTR + c * 4] = z;
        }
        for (int i = t; i < 8 * 49; i += 128) {       // K rows 24..31
            int r = 24 + i / 49, c = i % 49;
            *(float4*)&Ks[r * QSTR + c * 4] = z;
        }
        for (int i = t; i < 3 * 49; i += 128) {       // P rows 13..15
            int r = 13 + i / 49, c = i % 49;
            *(float4*)&Ps[r * QSTR + c * 4] = z;
        }
    }

    asm volatile("s_wait_asynccnt 0" ::: "memory");   // async transfers complete (per wave)
    __syncthreads();                                  // all waves' data visible

    const int wave = t >> 5;
    const int lane = t & 31;
    const int half = lane >> 4;     // 0: K-cols 0/1, 1: K-cols 2/3 (per 16x4 f32 layout)
    const int row  = lane & 15;     // M index for A, N index for B

    if (wave < 3) {
        // A fragment source: Q row `row`, contraction cols d0 + 2*half + {0,1}
        const float* arow = &Qs[row * QSTR + 2 * half];
        const float* brow;
        float* sdst; int scols, coff;
        if (wave == 0)      { brow = &Ks[row * QSTR + 2 * half];        sdst = Sac; scols = 32; coff = 0;  }
        else if (wave == 1) { brow = &Ks[(16 + row) * QSTR + 2 * half]; sdst = Sac; scols = 32; coff = 16; }
        else                { brow = &Ps[row * QSTR + 2 * half];        sdst = Sbd; scols = 16; coff = 0;  }

        v8f acc = {};
#pragma unroll
        for (int d0 = 0; d0 < Dc; d0 += 4) {
            v2f a, b;
            a.x = arow[d0]; a.y = arow[d0 + 1];
            b.x = brow[d0]; b.y = brow[d0 + 1];
            // D = A(16x4) * B(4x16) + C  -> v_wmma_f32_16x16x4_f32
            acc = __builtin_amdgcn_wmma_f32_16x16x4_f32(
                /*neg_a=*/false, a, /*neg_b=*/false, b,
                /*c_mod=*/(short)0, acc, /*reuse_a=*/false, /*reuse_b=*/false);
        }
        // C/D layout: VGPR r -> M = r + 8*half, N = row
#pragma unroll
        for (int r = 0; r < 8; ++r)
            sdst[(r + 8 * half) * scols + coff + row] = acc[r];
    }
    __syncthreads();

    // ---- relative shift + combine; out layout [B,H,W,U,K]; streaming NT stores ----
    const int b = bw / Wc, w = bw % Wc;
    float* obase = out + (size_t)(((b * Hc + h) * Wc + w)) * (Uc * Kc);
    for (int idx = t; idx < Uc * Kc; idx += 128) {
        int up = idx / Kc, c = idx % Kc;        // output (u', c)
        int u  = idx / (Kc + 1);                // source row in term_bd
        int f  = idx % (Kc + 1);                // source f (>=13 -> padded zero)
        float v = Sac[up * 32 + c];
        if (f < Fc) v += Sbd[u * 16 + f];
        __builtin_nontemporal_store(v, &obase[idx]);
    }
}

extern "C" void kernel_launch(void* const* d_in, const int* in_sizes, int n_in,
                              void* d_out, int out_size, void* d_ws, size_t ws_size,
                              hipStream_t stream) {
    const float* queries = (const float*)d_in[0];
    const float* keys    = (const float*)d_in[1];
    const float* pos_w   = (const float*)d_in[2];
    float* out = (float*)d_out;

    float* sinemb = (float*)d_ws;            // F*CH floats
    float* proj   = sinemb + Fc * CHc;       // F*CH floats

    const int n = Fc * CHc;                  // 19968
    sinemb_kernel<<<(n + 255) / 256, 256, 0, stream>>>(sinemb);
    proj_kernel  <<<(n + 255) / 256, 256, 0, stream>>>(sinemb, pos_w, proj);
    rel_attn_kernel<<<Bc * Wc * Hc, 128, 0, stream>>>(queries, keys, proj, out);
}